// CriterionOhem_146028888240
// MI455X (gfx1250) — compile-verified
//
#include <hip/hip_runtime.h>

#define BB 4
#define CC 19
#define HH 512
#define WW 1024
#define HWSZ (HH * WW)              // 524288 (power of two)
#define NPIX (BB * HWSZ)            // 2097152
#define TILE 256
#define NBIN1 16384
#define NBIN2 65536
#define MIN_KEPT_K 100000u
#define THRESH_F 0.7f
#define IGN 255
#define GRID1 1024
#define GRID5 1024

// One tile = 19 classes x 256 pixels x 4B = 19456 B = 1216 x 16B transfers.
// 1216 / 256 threads = 4.75 -> fixed 5 iterations/lane with the tail clamped
// to a duplicate (benign) transfer so EVERY wave issues exactly 5 async ops.
#define XFERS 1216
#define STAGE_ITERS 5

// ---------------------------------------------------------------------------
// Issue the async global->LDS stage of one tile (CDNA5 async engine,
// tracked by ASYNCcnt; 16B per lane per instruction).
//   base = pred + b*C*HW + rem_start   (tile-aligned, 16B-aligned segments)
//   sb   = LDS byte address of the destination buffer
// ---------------------------------------------------------------------------
__device__ __forceinline__ void stage_tile(const float* base, unsigned sb,
                                           int tid)
{
#pragma unroll
    for (int it = 0; it < STAGE_ITERS; ++it) {
        int k = tid + it * 256;
        k = (k < XFERS) ? k : (XFERS - 1);      // duplicate tail, same data
        const int c = k >> 6;                   // class row (64 xfers/row)
        const int j = k & 63;                   // 16B chunk within the row
        unsigned long long ga =
            (unsigned long long)(size_t)(base + (size_t)c * HWSZ + j * 4);
        unsigned la = sb + (unsigned)((c * TILE + j * 4) * 4);
        asm volatile("global_load_async_to_lds_b128 %0, %1, off"
                     :: "v"(la), "v"(ga) : "memory");
    }
}

// ---------------------------------------------------------------------------
// Pass 1: softmax / NLL / mask_prob per pixel + 16K-bin histogram of the
// top-16 bits of mask_prob's float pattern.  Tiles are staged into LDS by the
// async engine with a 2-deep ASYNCcnt-pipelined double buffer.
// ---------------------------------------------------------------------------
__global__ __launch_bounds__(256) void k_softmax_hist(
    const float* __restrict__ pred, const int* __restrict__ target,
    float* __restrict__ maskp, float* __restrict__ nllv,
    unsigned* __restrict__ hist1)
{
    __shared__ float    stage[2][CC * TILE];    // 2 x 19456 B
    __shared__ unsigned lhist[NBIN1];           // 64 KB block-local histogram

    const int tid = threadIdx.x;
    for (int i = tid; i < NBIN1; i += 256) lhist[i] = 0u;
    __syncthreads();

    const int ntile = NPIX / TILE;              // 8192 (uniform)

    // Prologue: stage first tile into buffer 0.
    {
        const int tile0 = blockIdx.x;           // < ntile always (1024 < 8192)
        const int pix0  = tile0 * TILE;
        const int b     = pix0 >> 19;
        const int rem   = pix0 & (HWSZ - 1);
        const float* base = pred + (size_t)b * ((size_t)CC * HWSZ) + rem;
        stage_tile(base, (unsigned)(size_t)(&stage[0][0]), tid);
    }

    int parity = 0;
    for (int tile = blockIdx.x; tile < ntile; tile += gridDim.x) {
        // Issue next tile's stage into the other buffer (uniform condition).
        const int nt = tile + gridDim.x;
        if (nt < ntile) {
            const int pixn = nt * TILE;
            const int bn   = pixn >> 19;
            const int remn = pixn & (HWSZ - 1);
            const float* basen =
                pred + (size_t)bn * ((size_t)CC * HWSZ) + remn;
            stage_tile(basen, (unsigned)(size_t)(&stage[parity ^ 1][0]), tid);
            // Only the 5 just-issued ops may remain in flight -> current
            // buffer's loads (issued earlier, in-order completion) are done.
            asm volatile("s_wait_asynccnt 5" ::: "memory");
        } else {
            asm volatile("s_wait_asynccnt 0" ::: "memory");
        }
        __syncthreads();                        // all waves' data visible

        const int pix = tile * TILE + tid;
        float x[CC];
        float xmax = -3.0e38f;
#pragma unroll
        for (int c = 0; c < CC; ++c) {
            x[c] = stage[parity][c * TILE + tid];
            xmax = fmaxf(xmax, x[c]);
        }

        const int  t     = target[pix];
        const bool valid = (t != IGN);
        const int  tc    = valid ? t : 0;

        float se = 0.0f;
        float xt = x[0];
#pragma unroll
        for (int c = 0; c < CC; ++c) {
            se += expf(x[c] - xmax);
            if (c == tc) xt = x[c];             // select, static index
        }
        const float lse  = xmax + logf(se);
        const float nll  = lse - xt;            // -log_softmax[target]
        const float prob = expf(xt - lse);      // softmax[target]
        const float m    = valid ? prob : 1.0f; // invalid sorts last

        maskp[pix] = m;
        nllv[pix]  = nll;
        atomicAdd(&lhist[__float_as_uint(m) >> 16], 1u);

        __syncthreads();                        // done reading buffer 'parity'
        parity ^= 1;
    }

    __syncthreads();
    for (int i = tid; i < NBIN1; i += 256) {
        const unsigned v = lhist[i];
        if (v) atomicAdd(&hist1[i], v);
    }
}

// ---------------------------------------------------------------------------
// Find the first-level bin containing the MIN_KEPT-th smallest value.
// ctrl[0] = bin index (top-16 bits), ctrl[1] = residual rank within bin.
// ---------------------------------------------------------------------------
__global__ void k_scan1(const unsigned* __restrict__ hist1,
                        unsigned* __restrict__ ctrl)
{
    __shared__ unsigned part[256];
    const int t = threadIdx.x;
    unsigned s = 0;
    for (int i = 0; i < NBIN1 / 256; ++i) s += hist1[t * (NBIN1 / 256) + i];
    part[t] = s;
    __syncthreads();
    if (t == 0) {
        unsigned cum = 0;
        int seg = 0;
        for (; seg < 256; ++seg) {
            if (cum + part[seg] >= MIN_KEPT_K) break;
            cum += part[seg];
        }
        int bin = seg * (NBIN1 / 256);
        for (;; ++bin) {
            const unsigned h = hist1[bin];
            if (cum + h >= MIN_KEPT_K) break;
            cum += h;
        }
        ctrl[0] = (unsigned)bin;
        ctrl[1] = MIN_KEPT_K - cum;   // 1-based rank within the bin
    }
}

// ---------------------------------------------------------------------------
// Second-level histogram: low-16 bits of candidates in the selected bin.
// ---------------------------------------------------------------------------
__global__ __launch_bounds__(256) void k_hist2(
    const float* __restrict__ maskp, const unsigned* __restrict__ ctrl,
    unsigned* __restrict__ hist2)
{
    const unsigned bin1 = ctrl[0];
    const int gsz = gridDim.x * blockDim.x;
    for (int i = blockIdx.x * blockDim.x + threadIdx.x; i < NPIX; i += gsz) {
        const unsigned bits = __float_as_uint(maskp[i]);
        if ((bits >> 16) == bin1) atomicAdd(&hist2[bits & 0xFFFFu], 1u);
    }
}

// ---------------------------------------------------------------------------
// Resolve exact kth bit pattern; threshold = max(kth, 0.7).  ctrl[2] = bits.
// ---------------------------------------------------------------------------
__global__ void k_scan2(const unsigned* __restrict__ hist2,
                        unsigned* __restrict__ ctrl)
{
    __shared__ unsigned part[256];
    const int t = threadIdx.x;
    unsigned s = 0;
    for (int i = 0; i < NBIN2 / 256; ++i) s += hist2[t * (NBIN2 / 256) + i];
    part[t] = s;
    __syncthreads();
    if (t == 0) {
        const unsigned rank = ctrl[1];
        unsigned cum = 0;
        int seg = 0;
        for (; seg < 256; ++seg) {
            if (cum + part[seg] >= rank) break;
            cum += part[seg];
        }
        int bin = seg * (NBIN2 / 256);
        for (;; ++bin) {
            const unsigned h = hist2[bin];
            if (cum + h >= rank) break;
            cum += h;
        }
        const unsigned kbits = (ctrl[0] << 16) | (unsigned)bin;
        const float thr = fmaxf(__uint_as_float(kbits), THRESH_F);
        ctrl[2] = __float_as_uint(thr);
    }
}

// ---------------------------------------------------------------------------
// Masked NLL sum + kept count; deterministic per-block LDS tree reduction.
// ---------------------------------------------------------------------------
__global__ __launch_bounds__(256) void k_reduce(
    const float* __restrict__ maskp, const float* __restrict__ nllv,
    const int* __restrict__ target, const unsigned* __restrict__ ctrl,
    float* __restrict__ partials)
{
    const float thr = __uint_as_float(ctrl[2]);
    float s = 0.0f, c = 0.0f;
    const int gsz = gridDim.x * blockDim.x;
    for (int i = blockIdx.x * blockDim.x + threadIdx.x; i < NPIX; i += gsz) {
        const int t = target[i];
        if (t != IGN && maskp[i] <= thr) { s += nllv[i]; c += 1.0f; }
    }
    __shared__ float rs[256], rc[256];
    rs[threadIdx.x] = s;
    rc[threadIdx.x] = c;
    __syncthreads();
    for (int off = 128; off > 0; off >>= 1) {
        if (threadIdx.x < (unsigned)off) {
            rs[threadIdx.x] += rs[threadIdx.x + off];
            rc[threadIdx.x] += rc[threadIdx.x + off];
        }
        __syncthreads();
    }
    if (threadIdx.x == 0) {
        partials[2 * blockIdx.x]     = rs[0];
        partials[2 * blockIdx.x + 1] = rc[0];
    }
}

__global__ void k_final(const float* __restrict__ partials,
                        float* __restrict__ out)
{
    __shared__ float rs[256], rc[256];
    float s = 0.0f, c = 0.0f;
    for (int i = threadIdx.x; i < GRID5; i += 256) {
        s += partials[2 * i];
        c += partials[2 * i + 1];
    }
    rs[threadIdx.x] = s;
    rc[threadIdx.x] = c;
    __syncthreads();
    for (int off = 128; off > 0; off >>= 1) {
        if (threadIdx.x < (unsigned)off) {
            rs[threadIdx.x] += rs[threadIdx.x + off];
            rc[threadIdx.x] += rc[threadIdx.x + off];
        }
        __syncthreads();
    }
    if (threadIdx.x == 0) out[0] = rs[0] / fmaxf(rc[0], 1.0f);
}

// ---------------------------------------------------------------------------
// Workspace layout (bytes):
//   [0,        65536)  hist1  u32[16384]
//   [65536,   327680)  hist2  u32[65536]
//   [327680,  327936)  ctrl   u32[...]: bin1, rank1, thr_bits
//   [327936,  336128)  partials f32[2*GRID5]
//   [524288,  524288+8M)    mask_prob f32[NPIX]
//   [524288+8M, 524288+16M) nll       f32[NPIX]
// Total ~17.3 MB.
// ---------------------------------------------------------------------------
extern "C" void kernel_launch(void* const* d_in, const int* in_sizes, int n_in,
                              void* d_out, int out_size, void* d_ws, size_t ws_size,
                              hipStream_t stream) {
    (void)in_sizes; (void)n_in; (void)out_size; (void)ws_size;
    const float* pred   = (const float*)d_in[0];
    const int*   target = (const int*)d_in[1];
    float*       out    = (float*)d_out;

    char* ws = (char*)d_ws;
    unsigned* hist1    = (unsigned*)(ws);
    unsigned* hist2    = (unsigned*)(ws + 65536);
    unsigned* ctrl     = (unsigned*)(ws + 327680);
    float*    partials = (float*)(ws + 327936);
    float*    maskp    = (float*)(ws + 524288);
    float*    nllv     = (float*)(ws + 524288 + (size_t)NPIX * 4);

    // zero histograms + ctrl (partials fully overwritten each launch)
    hipMemsetAsync(ws, 0, 524288, stream);

    k_softmax_hist<<<GRID1, 256, 0, stream>>>(pred, target, maskp, nllv, hist1);
    k_scan1<<<1, 256, 0, stream>>>(hist1, ctrl);
    k_hist2<<<GRID5, 256, 0, stream>>>(maskp, ctrl, hist2);
    k_scan2<<<1, 256, 0, stream>>>(hist2, ctrl);
    k_reduce<<<GRID5, 256, 0, stream>>>(maskp, nllv, target, ctrl, partials);
    k_final<<<1, 256, 0, stream>>>(partials, out);
}